// CustomGCN_22909355557151
// MI455X (gfx1250) — compile-verified
//
#include <hip/hip_runtime.h>

#define N_NODES 100000
#define E_EDGES 3200000
#define IN_C    512
#define HID     16
#define OUT_C   64

typedef __attribute__((ext_vector_type(2))) float v2f;
typedef __attribute__((ext_vector_type(8))) float v8f;

// ---------------------------------------------------------------------------
// Zero a float region (accumulators must start at 0 on every replay).
// ---------------------------------------------------------------------------
__global__ void zero_kernel(float* __restrict__ p, int n) {
    int i = blockIdx.x * blockDim.x + threadIdx.x;
    if (i < n) p[i] = 0.0f;
}

// ---------------------------------------------------------------------------
// conv2 algebraic collapse: consts = {sum(W2^2), sum(W2*b2), sum(b2^2)}
// ---------------------------------------------------------------------------
__global__ void w2_reduce_kernel(const float* __restrict__ W2,
                                 const float* __restrict__ b2,
                                 float* __restrict__ consts) {
    __shared__ float sa[OUT_C], sb[OUT_C], sc[OUT_C];
    int t = threadIdx.x;
    float w = W2[t], b = b2[t];
    sa[t] = w * w; sb[t] = w * b; sc[t] = b * b;
    __syncthreads();
    if (t == 0) {
        float A = 0.f, B = 0.f, C = 0.f;
        for (int i = 0; i < OUT_C; ++i) { A += sa[i]; B += sb[i]; C += sc[i]; }
        consts[0] = A; consts[1] = B; consts[2] = C;
    }
}

// ---------------------------------------------------------------------------
// h = x @ W1 + b1   via V_WMMA_F32_16X16X4_F32.
// One 16-row tile per wave, 8 waves (256 threads) per block.
// W1 staged in LDS, pre-swizzled into per-kstep B fragments:
//   B frag layout (4x16, 32-bit): lanes 0-15 hold K={0,1}, lanes 16-31 K={2,3}
// A frag layout (16x4, 32-bit):   lanes 0-15 rows M, K={0,1}; lanes 16-31 K={2,3}
// ---------------------------------------------------------------------------
__global__ __launch_bounds__(256) void gemm_h_kernel(const float* __restrict__ x,
                                                     const float* __restrict__ W1,
                                                     const float* __restrict__ b1,
                                                     float* __restrict__ h,
                                                     int nTiles) {
    __shared__ float sB[(IN_C / 4) * 64];   // 128 ksteps * 64 floats = 32 KB
    __shared__ float sb1[HID];

    // Stage W1 (coalesced read, swizzled write so B fragment is a contiguous b64)
    for (int i = threadIdx.x; i < IN_C * HID; i += blockDim.x) {
        int k = i / HID, n = i % HID;
        int kstep = k >> 2, koff = k & 3;
        int lane  = ((koff >> 1) << 4) | n;   // which lane owns it
        int slot  = koff & 1;                 // which of the 2 VGPRs
        sB[kstep * 64 + lane * 2 + slot] = W1[i];
    }
    if (threadIdx.x < HID) sb1[threadIdx.x] = b1[threadIdx.x];
    __syncthreads();

    const int wave = threadIdx.x >> 5;
    const int lane = threadIdx.x & 31;
    const int tile = blockIdx.x * 8 + wave;
    if (tile >= nTiles) return;               // whole-wave exit: EXEC all-1 for WMMA

    const int row = tile * 16 + (lane & 15);
    const float* aptr = x + (size_t)row * IN_C + ((lane >> 4) << 1);

    v8f c = {};
    for (int ko = 0; ko < IN_C / 4; ko += 16) {
        // speculative prefetch of the A stream one chunk ahead (gfx1250 global_prefetch_b8)
        __builtin_prefetch(aptr + (ko + 16) * 4, 0, 0);
#pragma unroll
        for (int ki = 0; ki < 16; ++ki) {
            const int ks = ko + ki;
            v2f a = *(const v2f*)(aptr + ks * 4);
            v2f b = *(const v2f*)(&sB[ks * 64 + lane * 2]);
            c = __builtin_amdgcn_wmma_f32_16x16x4_f32(
                    /*neg_a=*/false, a, /*neg_b=*/false, b,
                    /*c_mod=*/(short)0, c, /*reuse_a=*/false, /*reuse_b=*/false);
        }
    }

    // D layout: VGPR r -> M = r (lanes 0-15) / M = r+8 (lanes 16-31), N = lane&15
    const int col   = lane & 15;
    const int rbase = tile * 16 + ((lane >> 4) << 3);
    const float bias = sb1[col];
#pragma unroll
    for (int r = 0; r < 8; ++r) {
        h[(size_t)(rbase + r) * HID + col] = c[r] + bias;
    }
}

// ---------------------------------------------------------------------------
// conv1 edge phase: m = sigmoid(<h[dst], h[src]>); scatter-sum to s, count to cnt.
// One thread per edge (incl. self-loops for e >= E).
// ---------------------------------------------------------------------------
__global__ void edge1_kernel(const int* __restrict__ ei,
                             const float* __restrict__ h,
                             float* __restrict__ s,
                             float* __restrict__ cnt,
                             int total) {
    int e = blockIdx.x * blockDim.x + threadIdx.x;
    if (e >= total) return;
    int src, dst;
    if (e < E_EDGES) { src = ei[e]; dst = ei[E_EDGES + e]; }
    else             { src = dst = e - E_EDGES; }

    const float4* hs = (const float4*)(h + (size_t)src * HID);
    const float4* hd = (const float4*)(h + (size_t)dst * HID);
    float dot = 0.f;
#pragma unroll
    for (int i = 0; i < 4; ++i) {
        float4 a = hs[i], b = hd[i];
        dot += a.x * b.x + a.y * b.y + a.z * b.z + a.w * b.w;
    }
    float m = 1.0f / (1.0f + expf(-dot));
    atomicAdd(&s[dst], m);
    atomicAdd(&cnt[dst], 1.0f);
}

// ---------------------------------------------------------------------------
// finish conv1: h1 = relu(mean)
// ---------------------------------------------------------------------------
__global__ void node1_kernel(const float* __restrict__ s,
                             const float* __restrict__ cnt,
                             float* __restrict__ h1, int n) {
    int i = blockIdx.x * blockDim.x + threadIdx.x;
    if (i >= n) return;
    float v = s[i] / fmaxf(cnt[i], 1.0f);
    h1[i] = fmaxf(v, 0.0f);
}

// ---------------------------------------------------------------------------
// conv2 edge phase, collapsed: dot = a*b*A2 + (a+b)*B2 + C2  (a=h1[src], b=h1[dst])
// ---------------------------------------------------------------------------
__global__ void edge2_kernel(const int* __restrict__ ei,
                             const float* __restrict__ h1,
                             const float* __restrict__ consts,
                             float* __restrict__ s2,
                             float* __restrict__ cnt2,
                             int total) {
    int e = blockIdx.x * blockDim.x + threadIdx.x;
    if (e >= total) return;
    int src, dst;
    if (e < E_EDGES) { src = ei[e]; dst = ei[E_EDGES + e]; }
    else             { src = dst = e - E_EDGES; }

    float a = h1[src], b = h1[dst];
    float dot = a * b * consts[0] + (a + b) * consts[1] + consts[2];
    float m = 1.0f / (1.0f + expf(-dot));
    atomicAdd(&s2[dst], m);
    atomicAdd(&cnt2[dst], 1.0f);
}

// ---------------------------------------------------------------------------
// final: log_softmax over a width-1 axis == v - v == 0 (kept data-dependent)
// ---------------------------------------------------------------------------
__global__ void out_kernel(const float* __restrict__ s2,
                           const float* __restrict__ cnt2,
                           float* __restrict__ out, int n) {
    int i = blockIdx.x * blockDim.x + threadIdx.x;
    if (i >= n) return;
    float v = s2[i] / fmaxf(cnt2[i], 1.0f);
    out[i] = v - v;
}

extern "C" void kernel_launch(void* const* d_in, const int* in_sizes, int n_in,
                              void* d_out, int out_size, void* d_ws, size_t ws_size,
                              hipStream_t stream) {
    const float* x  = (const float*)d_in[0];
    const int*   ei = (const int*)d_in[1];
    const float* W1 = (const float*)d_in[2];
    const float* b1 = (const float*)d_in[3];
    const float* W2 = (const float*)d_in[4];
    const float* b2 = (const float*)d_in[5];

    float* ws     = (float*)d_ws;
    float* h      = ws;                                  // N*16
    float* s      = h  + (size_t)N_NODES * HID;          // N
    float* cnt    = s  + N_NODES;                        // N
    float* h1     = cnt + N_NODES;                       // N
    float* s2     = h1 + N_NODES;                        // N
    float* cnt2   = s2 + N_NODES;                        // N
    float* consts = cnt2 + N_NODES;                      // 3

    const int total = E_EDGES + N_NODES;                 // edges + self-loops

    // zero all accumulators (s..cnt2 + consts) every launch
    const int zn = 5 * N_NODES + 3;
    zero_kernel<<<(zn + 255) / 256, 256, 0, stream>>>(s, zn);

    w2_reduce_kernel<<<1, OUT_C, 0, stream>>>(W2, b2, consts);

    const int nTiles = N_NODES / 16;                     // 6250, exact
    gemm_h_kernel<<<(nTiles + 7) / 8, 256, 0, stream>>>(x, W1, b1, h, nTiles);

    edge1_kernel<<<(total + 255) / 256, 256, 0, stream>>>(ei, h, s, cnt, total);
    node1_kernel<<<(N_NODES + 255) / 256, 256, 0, stream>>>(s, cnt, h1, N_NODES);
    edge2_kernel<<<(total + 255) / 256, 256, 0, stream>>>(ei, h1, consts, s2, cnt2, total);
    out_kernel<<<(out_size + 255) / 256, 256, 0, stream>>>(s2, cnt2, (float*)d_out, out_size);
}